// MPSN_L_29257317220559
// MI455X (gfx1250) — compile-verified
//
#include <hip/hip_runtime.h>
#include <hip/hip_bf16.h>

#define NNODES 100000
#define EDGES  3200000
#define FDIM   128
#define ODIM   64

typedef float v2f __attribute__((ext_vector_type(2)));
typedef float v8f __attribute__((ext_vector_type(8)));

// ---------------------------------------------------------------------------
// Utility: zero an int buffer (grid-stride).
// ---------------------------------------------------------------------------
__global__ void zero_int_kernel(int* __restrict__ p, int n) {
    int i = blockIdx.x * blockDim.x + threadIdx.x;
    int stride = gridDim.x * blockDim.x;
    for (; i < n; i += stride) p[i] = 0;
}

// ---------------------------------------------------------------------------
// CSR build step 1: row histogram (native u32 atomics, ~3.2M total).
// ---------------------------------------------------------------------------
__global__ void hist_kernel(const int* __restrict__ rows, int* __restrict__ counts, int n) {
    int i = blockIdx.x * blockDim.x + threadIdx.x;
    int stride = gridDim.x * blockDim.x;
    for (; i < n; i += stride) atomicAdd(&counts[rows[i]], 1);
}

// ---------------------------------------------------------------------------
// CSR build step 2: exclusive prefix scan of counts[0..n) -> rowptr[0..n].
// Single 1024-thread block, LDS Hillis-Steele per 1024-tile + running carry.
// 100k elements => ~98 tiles; microseconds of work.
// ---------------------------------------------------------------------------
__global__ __launch_bounds__(1024) void scan_kernel(const int* __restrict__ counts,
                                                    int* __restrict__ rowptr, int n) {
    __shared__ int buf[1024];
    __shared__ int carry;
    if (threadIdx.x == 0) carry = 0;
    __syncthreads();
    for (int base = 0; base < n; base += 1024) {
        int idx = base + (int)threadIdx.x;
        int v = (idx < n) ? counts[idx] : 0;
        buf[threadIdx.x] = v;
        __syncthreads();
        #pragma unroll
        for (int off = 1; off < 1024; off <<= 1) {
            int t = (threadIdx.x >= off) ? buf[threadIdx.x - off] : 0;
            __syncthreads();
            buf[threadIdx.x] += t;
            __syncthreads();
        }
        int excl = buf[threadIdx.x] - v;
        int total = buf[1023];
        if (idx < n) rowptr[idx] = carry + excl;
        __syncthreads();
        if (threadIdx.x == 0) carry += total;
        __syncthreads();
    }
    if (threadIdx.x == 0) rowptr[n] = carry;
}

// ---------------------------------------------------------------------------
// CSR build step 3: cursor = rowptr[0..n).
// ---------------------------------------------------------------------------
__global__ void copy_int_kernel(const int* __restrict__ src, int* __restrict__ dst, int n) {
    int i = blockIdx.x * blockDim.x + threadIdx.x;
    int stride = gridDim.x * blockDim.x;
    for (; i < n; i += stride) dst[i] = src[i];
}

// ---------------------------------------------------------------------------
// CSR build step 4: scatter edges into row-sorted (col,val) arrays.
// ---------------------------------------------------------------------------
__global__ void scatter_kernel(const int* __restrict__ rows, const int* __restrict__ cols,
                               const float* __restrict__ vals, int* __restrict__ cursor,
                               int* __restrict__ scols, float* __restrict__ svals, int n) {
    int i = blockIdx.x * blockDim.x + threadIdx.x;
    int stride = gridDim.x * blockDim.x;
    for (; i < n; i += stride) {
        int p = atomicAdd(&cursor[rows[i]], 1);
        scols[p] = cols[i];
        svals[p] = vals[i];
    }
}

// ---------------------------------------------------------------------------
// CSR SpMM: out[r,:] = sum_e val[e] * H[col[e],:], one row per wave32.
// Lane i owns features [4i,4i+4): coalesced 512B float4 gathers from the
// L2-resident H (51.2MB << 192MB L2), register accumulation, single plain
// float4 store per row — no output atomics at all. Unrolled x2 so two
// gathers are in flight per iteration.
// ---------------------------------------------------------------------------
__global__ __launch_bounds__(256) void spmm_csr_kernel(
    const int* __restrict__ rowptr, const int* __restrict__ scols,
    const float* __restrict__ svals, const float* __restrict__ H,
    float* __restrict__ out, int nrows)
{
    int wid  = (int)((blockIdx.x * blockDim.x + threadIdx.x) >> 5);
    int lane = threadIdx.x & 31;
    if (wid >= nrows) return;
    int beg = rowptr[wid];
    int end = rowptr[wid + 1];
    float4 acc = {0.f, 0.f, 0.f, 0.f};
    int i = beg;
    for (; i + 1 < end; i += 2) {
        int   c0 = scols[i],     c1 = scols[i + 1];
        float v0 = svals[i],     v1 = svals[i + 1];
        float4 h0 = ((const float4*)(H + (size_t)c0 * FDIM))[lane];
        float4 h1 = ((const float4*)(H + (size_t)c1 * FDIM))[lane];
        acc.x += v0 * h0.x + v1 * h1.x;
        acc.y += v0 * h0.y + v1 * h1.y;
        acc.z += v0 * h0.z + v1 * h1.z;
        acc.w += v0 * h0.w + v1 * h1.w;
    }
    if (i < end) {
        int   c = scols[i];
        float v = svals[i];
        float4 h = ((const float4*)(H + (size_t)c * FDIM))[lane];
        acc.x += v * h.x; acc.y += v * h.y; acc.z += v * h.z; acc.w += v * h.w;
    }
    ((float4*)(out + (size_t)wid * FDIM))[lane] = acc;
}

// ---------------------------------------------------------------------------
// Fused multi-GEMM + tanh epilogue, full-precision f32 WMMA (verified lowering
// to v_wmma_f32_16x16x4_f32):  out = act( A0@W0 [+ A1@W1 + A2@W2] ).
// One wave per 16x16 output tile; block = (M/16) waves sharing 16 A-rows.
// ISA VGPR layouts (wave32):
//   A 16x4 : lane L -> (M = L%16, K = 2*(L/16) + {0,1})
//   B 4x16 : lane L -> (N = L%16, K = 2*(L/16) + {0,1})
//   C 16x16: VGPR r -> (M = r + 8*(L/16), N = L%16)
// ---------------------------------------------------------------------------
template<int NMAT, bool DO_TANH>
__global__ void gemm_wmma_kernel(
    const float* __restrict__ A0, const float* __restrict__ W0,
    const float* __restrict__ A1, const float* __restrict__ W1,
    const float* __restrict__ A2, const float* __restrict__ W2,
    float* __restrict__ out, int K, int M)
{
    const int lane = threadIdx.x & 31;
    const int wave = threadIdx.x >> 5;
    const int m    = lane & 15;
    const int kh   = lane >> 4;
    const int n    = wave * 16 + (lane & 15);
    const long rowA = (long)blockIdx.x * 16 + m;

    const float* a0p = A0 + rowA * K;
    const float* a1p = (NMAT > 1) ? (A1 + rowA * K) : nullptr;
    const float* a2p = (NMAT > 2) ? (A2 + rowA * K) : nullptr;

    v8f acc = {};
    for (int k = 0; k < K; k += 4) {
        const int ka = k + kh * 2;
        v2f a, b;

        a = *(const v2f*)(a0p + ka);
        b.x = W0[(size_t)(ka + 0) * M + n];
        b.y = W0[(size_t)(ka + 1) * M + n];
        acc = __builtin_amdgcn_wmma_f32_16x16x4_f32(false, a, false, b,
                                                    (short)0, acc, false, false);
        if constexpr (NMAT > 1) {
            a = *(const v2f*)(a1p + ka);
            b.x = W1[(size_t)(ka + 0) * M + n];
            b.y = W1[(size_t)(ka + 1) * M + n];
            acc = __builtin_amdgcn_wmma_f32_16x16x4_f32(false, a, false, b,
                                                        (short)0, acc, false, false);
        }
        if constexpr (NMAT > 2) {
            a = *(const v2f*)(a2p + ka);
            b.x = W2[(size_t)(ka + 0) * M + n];
            b.y = W2[(size_t)(ka + 1) * M + n];
            acc = __builtin_amdgcn_wmma_f32_16x16x4_f32(false, a, false, b,
                                                        (short)0, acc, false, false);
        }
    }

    const long obase = (long)blockIdx.x * 16 + kh * 8;
    #pragma unroll
    for (int r = 0; r < 8; ++r) {
        float v = acc[r];
        if constexpr (DO_TANH) v = tanhf(v);
        out[(obase + r) * M + n] = v;
    }
}

// ---------------------------------------------------------------------------
// Row-wise L2 normalize (eps clamp) + tanh, one wave32 per 64-wide row.
// ---------------------------------------------------------------------------
__global__ __launch_bounds__(256) void norm_tanh_kernel(float* __restrict__ y, int nrows) {
    int wid  = (int)((blockIdx.x * blockDim.x + threadIdx.x) >> 5);
    int lane = threadIdx.x & 31;
    if (wid >= nrows) return;
    float* row = y + (size_t)wid * ODIM;
    float v0 = row[lane];
    float v1 = row[lane + 32];
    float ss = v0 * v0 + v1 * v1;
    #pragma unroll
    for (int off = 16; off > 0; off >>= 1)
        ss += __shfl_xor(ss, off, 32);
    float inv = 1.0f / fmaxf(sqrtf(ss), 1e-12f);
    row[lane]      = tanhf(v0 * inv);
    row[lane + 32] = tanhf(v1 * inv);
}

// ---------------------------------------------------------------------------
extern "C" void kernel_launch(void* const* d_in, const int* in_sizes, int n_in,
                              void* d_out, int out_size, void* d_ws, size_t ws_size,
                              hipStream_t stream) {
    const float* X    = (const float*)d_in[0];
    const int*   Lu_r = (const int*)  d_in[1];
    const int*   Lu_c = (const int*)  d_in[2];
    const float* Lu_v = (const float*)d_in[3];
    const int*   Ld_r = (const int*)  d_in[4];
    const int*   Ld_c = (const int*)  d_in[5];
    const float* Ld_v = (const float*)d_in[6];
    const float* Ws1 = (const float*)d_in[7],  *Wu1 = (const float*)d_in[8],  *Wd1 = (const float*)d_in[9];
    const float* Ws2 = (const float*)d_in[10], *Wu2 = (const float*)d_in[11], *Wd2 = (const float*)d_in[12];
    const float* Ws3 = (const float*)d_in[13], *Wu3 = (const float*)d_in[14], *Wd3 = (const float*)d_in[15];
    const float* Wfc = (const float*)d_in[16];
    float* out = (float*)d_out;

    // ---- workspace carve-up -------------------------------------------------
    const size_t NF = (size_t)NNODES * FDIM;
    float* UP = (float*)d_ws;           // [N,128]
    float* DN = UP + NF;                // [N,128]
    float* HA = DN + NF;                // [N,128] ping
    float* HB = HA + NF;                // [N,128] pong
    int* rowptr_u = (int*)(HB + NF);    // N+1
    int* rowptr_d = rowptr_u + (NNODES + 2);     // N+1 (keep 8B-ish spacing)
    int* cnt_u    = rowptr_d + (NNODES + 2);     // N (histogram, then cursor)
    int* cnt_d    = cnt_u + NNODES;              // N
    int* scols_u  = cnt_d + NNODES;              // E
    int* scols_d  = scols_u + EDGES;             // E
    float* svals_u = (float*)(scols_d + EDGES);  // E
    float* svals_d = svals_u + EDGES;            // E

    const int GS_BLOCKS = 2048;                  // grid-stride kernels
    const int row_tiles = NNODES / 16;           // 6250 exactly
    const int wave_rows = (NNODES + 7) / 8;      // one wave per row, 8 waves/blk

    // ---- build CSR for both Laplacians (reused by all 3 layers) ------------
    auto build_csr = [&](const int* r, const int* c, const float* v,
                         int* rowptr, int* cnt, int* scols, float* svals) {
        zero_int_kernel<<<256, 256, 0, stream>>>(cnt, NNODES);
        hist_kernel<<<GS_BLOCKS, 256, 0, stream>>>(r, cnt, EDGES);
        scan_kernel<<<1, 1024, 0, stream>>>(cnt, rowptr, NNODES);
        copy_int_kernel<<<256, 256, 0, stream>>>(rowptr, cnt, NNODES);  // cursor
        scatter_kernel<<<GS_BLOCKS, 256, 0, stream>>>(r, c, v, cnt, scols, svals, EDGES);
    };
    build_csr(Lu_r, Lu_c, Lu_v, rowptr_u, cnt_u, scols_u, svals_u);
    build_csr(Ld_r, Ld_c, Ld_v, rowptr_d, cnt_d, scols_d, svals_d);

    // ---- 3 message-passing layers ------------------------------------------
    auto layer = [&](const float* Hin, const float* Ws, const float* Wu,
                     const float* Wd, float* Hout, int K, int M) {
        spmm_csr_kernel<<<wave_rows, 256, 0, stream>>>(rowptr_u, scols_u, svals_u, Hin, UP, NNODES);
        spmm_csr_kernel<<<wave_rows, 256, 0, stream>>>(rowptr_d, scols_d, svals_d, Hin, DN, NNODES);
        gemm_wmma_kernel<3, true><<<row_tiles, (M / 16) * 32, 0, stream>>>(
            Hin, Ws, UP, Wu, DN, Wd, Hout, K, M);
    };

    layer(X,  Ws1, Wu1, Wd1, HA, 128, 128);
    layer(HA, Ws2, Wu2, Wd2, HB, 128, 128);
    layer(HB, Ws3, Wu3, Wd3, HA, 128, 64);   // HA now holds [N,64] h3

    // ---- FC + normalize + tanh ---------------------------------------------
    gemm_wmma_kernel<1, false><<<row_tiles, (ODIM / 16) * 32, 0, stream>>>(
        HA, Wfc, nullptr, nullptr, nullptr, nullptr, out, ODIM, ODIM);
    norm_tanh_kernel<<<(NNODES + 7) / 8, 256, 0, stream>>>(out, NNODES);
}